// NeuralSheet_85375359910332
// MI455X (gfx1250) — compile-verified
//
#include <hip/hip_runtime.h>
#include <cmath>

#define S 128
#define IN_DIM 96
#define RF 17
#define W 25
#define WW (W*W)            // 625
#define HALF (W/2)          // 12
#define AFF 0.5f
#define ITERS 30
#define PADW (S + W - 1)    // 152
#define NPIX (S*S)          // 16384
#define TPR 7               // WMMA steps per dy-row (K row padded 25 -> 28)

typedef float v2f __attribute__((ext_vector_type(2)));
typedef float v8f __attribute__((ext_vector_type(8)));

// One-time: afferent pass -> net_aff = (afferent - thresholds)*AFF ; also zero resp0.
__global__ __launch_bounds__(256) void prep_kernel(
    const float* __restrict__ img, const float* __restrict__ aw,
    const float* __restrict__ thr, const int* __restrict__ row_idx,
    const int* __restrict__ col_idx, float* __restrict__ net_aff,
    float* __restrict__ resp0)
{
    int n = blockIdx.x * 256 + threadIdx.x;          // 64 blocks x 256 = 16384
    int i = n >> 7, j = n & 127;
    const float* wp = aw + (size_t)n * (RF * RF);
    float acc = 0.f;
    for (int r = 0; r < RF; ++r) {
        int ri = row_idx[i * RF + r];
        const float* imrow = img + ri * IN_DIM;
        for (int c = 0; c < RF; ++c) {
            int cj = col_idx[j * RF + c];
            acc = fmaf(imrow[cj], wp[r * RF + c], acc);
        }
    }
    net_aff[n] = (acc - thr[n]) * AFF;
    resp0[n] = 0.f;
}

// One-time: gmean = mean(gains)
__global__ __launch_bounds__(256) void gmean_kernel(
    const float* __restrict__ gains, float* __restrict__ gmean)
{
    __shared__ float red[256];
    int tid = threadIdx.x;
    float s = 0.f;
    for (int k = tid; k < NPIX; k += 256) s += gains[k];
    red[tid] = s;
    __syncthreads();
    for (int off = 128; off > 0; off >>= 1) {
        if (tid < off) red[tid] += red[tid + off];
        __syncthreads();
    }
    if (tid == 0) gmean[0] = red[0] / (float)NPIX;
}

// Per-iteration: one block per image row i; 8 waves, each wave owns 16 pixels
// (j0..j0+15): 625-term lateral dots = diag(A x B^T) via v_wmma_f32_16x16x4_f32.
// K reordered row-major, each 25-wide dy-row padded to 28 -> 7 steps/row, the
// 7th step masked branchlessly (amask). No offset table, no dependent LDS chain.
__global__ __launch_bounds__(256) void iter_kernel(
    const float* __restrict__ resp_in, float* __restrict__ resp_out,
    const float* __restrict__ net_aff, const float* __restrict__ gmean,
    const float* __restrict__ crops)
{
    __shared__ float tile[W * PADW];                  // 25 x 152 padded resp rows
    const int i   = blockIdx.x;
    const int tid = threadIdx.x;

    // Stage rows i-12..i+12 of resp, zero-padded 12 each side (row-wise, no division).
    if (tid < PADW) {
        int gx = tid - HALF;
        bool xok = (gx >= 0) && (gx < S);
        for (int r = 0; r < W; ++r) {
            int gy = i + r - HALF;
            float v = 0.f;
            if (xok && gy >= 0 && gy < S) v = resp_in[gy * S + gx];
            tile[r * PADW + tid] = v;
        }
    }
    __syncthreads();

    const int wave = tid >> 5;
    const int lane = tid & 31;
    const int j0   = wave * 16;
    const int m    = lane & 15;                      // A row / B col for this lane
    const int kh   = lane >> 4;                      // half-wave -> K offset 0 or 2
    const int j    = j0 + m;
    const int n    = i * S + j;
    const int ib   = j + 2 * kh;                     // lane's base column in tile
    const float gm = gmean[0];
    const float* crops_n  = crops + n;               // + k*NPIX per plane
    const float* crops_nk = crops_n + (size_t)(2 * kh) * NPIX;
    const float amask = (kh == 0) ? 1.f : 0.f;       // masks the dx=24..27 step

#if __has_builtin(__builtin_amdgcn_wmma_f32_16x16x4_f32)
    v8f c = {0.f, 0.f, 0.f, 0.f, 0.f, 0.f, 0.f, 0.f};

    v2f a0[TPR], b0[TPR], a1[TPR], b1[TPR];

    // Load all 7 steps of dy-row `dy` into (ra, rb).
    auto load_row = [&](int dy, v2f* ra, v2f* rb) {
        const float* cr = crops_nk + (size_t)(dy * W) * NPIX;   // k = dy*25 + 2kh
        const float* tr = tile + dy * PADW + ib;                // immediates off one base
#pragma unroll
        for (int t = 0; t < TPR - 1; ++t) {          // dx = 4t+2kh+{0,1} <= 23: in-bounds
            ra[t].x = cr[(size_t)(4 * t) * NPIX];
            ra[t].y = cr[(size_t)(4 * t + 1) * NPIX];
            rb[t].x = tr[4 * t];
            rb[t].y = tr[4 * t + 1];
            __builtin_prefetch(&cr[(size_t)(4 * t + W) * NPIX], 0, 1); // next row ahead
        }
        // Step 6: only dx=24 (kh==0, comp 0) is real; mask A, leave B raw.
        ra[TPR-1].x = crops_n[(size_t)(dy * W + (W - 1)) * NPIX] * amask;
        ra[TPR-1].y = 0.f;
        rb[TPR-1].x = tile[dy * PADW + (W - 1) + j];
        rb[TPR-1].y = 0.f;
    };

    auto wmma_row = [&](const v2f* ra, const v2f* rb) {
#pragma unroll
        for (int t = 0; t < TPR; ++t)
            c = __builtin_amdgcn_wmma_f32_16x16x4_f32(
                false, ra[t], false, rb[t], (short)0, c, false, false);
    };

    // Row-granular ping-pong: consume row dy while loading row dy+1.
    load_row(0, a0, b0);
    for (int dy = 0; dy < W - 1; dy += 2) {          // dy = 0,2,...,22
        load_row(dy + 1, a1, b1);
        wmma_row(a0, b0);
        load_row(dy + 2, a0, b0);
        wmma_row(a1, b1);
    }
    wmma_row(a0, b0);                                // row 24

    // Diagonal: pixel p (0..7) -> lane p comp p ; pixel p (8..15) -> lane p+16 comp p-8.
    int s = lane & 7;
    float dv = c[0];
    if (s == 1) dv = c[1];
    if (s == 2) dv = c[2];
    if (s == 3) dv = c[3];
    if (s == 4) dv = c[4];
    if (s == 5) dv = c[5];
    if (s == 6) dv = c[6];
    if (s == 7) dv = c[7];
    bool active = (lane < 8) || (lane >= 24);
    int pix = (lane < 8) ? lane : (lane - 16);
    if (active) {
        int no = i * S + j0 + pix;
        float x = (net_aff[no] + dv * (1.f - AFF)) * gm;
        x = fmaxf(x, 0.f);
        resp_out[no] = tanhf(x);
    }
#else
    // VALU fallback (fp32-exact): lanes 0..15 of each wave each own one pixel.
    if (kh == 0) {
        float acc = 0.f;
        for (int dy = 0; dy < W; ++dy)
            for (int dx = 0; dx < W; ++dx)
                acc = fmaf(crops_n[(size_t)(dy * W + dx) * NPIX],
                           tile[dy * PADW + dx + j], acc);
        float x = (net_aff[n] + acc * (1.f - AFF)) * gm;
        x = fmaxf(x, 0.f);
        resp_out[n] = tanhf(x);
    }
#endif
}

extern "C" void kernel_launch(void* const* d_in, const int* in_sizes, int n_in,
                              void* d_out, int out_size, void* d_ws, size_t ws_size,
                              hipStream_t stream) {
    const float* img   = (const float*)d_in[0];   // (1,1,96,96)
    const float* aw    = (const float*)d_in[1];   // (16384,17,17)
    const float* crops = (const float*)d_in[2];   // (625,128,128)
    const float* thr   = (const float*)d_in[3];   // (128,128)
    const float* gains = (const float*)d_in[4];   // (128,128)
    const int*   ridx  = (const int*)d_in[5];     // (128,17)
    const int*   cidx  = (const int*)d_in[6];     // (128,17)
    float* out = (float*)d_out;                   // (1,1,128,128)

    float* w       = (float*)d_ws;
    float* net_aff = w;                 // 16384 f
    float* gmean   = w + NPIX;          // 1 f (padded)
    float* respA   = w + NPIX + 16;     // 16384 f
    float* respB   = respA + NPIX;      // 16384 f

    prep_kernel<<<NPIX / 256, 256, 0, stream>>>(img, aw, thr, ridx, cidx, net_aff, respA);
    gmean_kernel<<<1, 256, 0, stream>>>(gains, gmean);

    float* bufs[2] = {respA, respB};
    for (int t = 0; t < ITERS; ++t) {
        const float* in = bufs[t & 1];
        float* o = (t == ITERS - 1) ? out : bufs[(t + 1) & 1];
        iter_kernel<<<S, 256, 0, stream>>>(in, o, net_aff, gmean, crops);
    }
}